// CenterLoss_90366111908363
// MI455X (gfx1250) — compile-verified
//
#include <hip/hip_runtime.h>
#include <math.h>

// Problem constants (from reference)
#define B_  1024
#define C_  50000
#define D_  256
#define LR_ 0.5f

typedef __attribute__((ext_vector_type(2))) float v2f;
typedef __attribute__((ext_vector_type(4))) float v4f;
typedef __attribute__((ext_vector_type(8))) float v8f;

// ---------------------------------------------------------------------------
// K0: zero per-class counters
// ---------------------------------------------------------------------------
__global__ void k_zero_counts(int* __restrict__ counts) {
    int c = blockIdx.x * blockDim.x + threadIdx.x;
    if (c < C_) counts[c] = 0;
}

// ---------------------------------------------------------------------------
// K1: new_center starts as a copy of center (float4, grid-stride).
// Regular (RT) loads on purpose: warms L2 with center for the gather/scatter
// kernels that follow (center = 51.2 MB, fits the 192 MB L2 easily).
// ---------------------------------------------------------------------------
__global__ void k_copy_center(const v4f* __restrict__ src,
                              v4f* __restrict__ dst, int n4) {
    int i = blockIdx.x * blockDim.x + threadIdx.x;
    int stride = gridDim.x * blockDim.x;
    for (; i < n4; i += stride) dst[i] = src[i];
}

// ---------------------------------------------------------------------------
// K2: scan one-hot rows of y -> labels[b], counts[c]. One block per row,
// float4-coalesced (C = 50000 divisible by 4). 204.8 MB read-once stream:
// non-temporal loads keep it from evicting center out of L2.
// ---------------------------------------------------------------------------
__global__ void k_find_labels(const float* __restrict__ y,
                              int* __restrict__ labels,
                              int* __restrict__ counts) {
    int b = blockIdx.x;
    const v4f* row = (const v4f*)(y + (size_t)b * C_);
    for (int c4 = threadIdx.x; c4 < C_ / 4; c4 += blockDim.x) {
        v4f v = __builtin_nontemporal_load(&row[c4]);
        int base = c4 * 4;
        if (v.x > 0.5f) { labels[b] = base + 0; atomicAdd(&counts[base + 0], 1); }
        if (v.y > 0.5f) { labels[b] = base + 1; atomicAdd(&counts[base + 1], 1); }
        if (v.z > 0.5f) { labels[b] = base + 2; atomicAdd(&counts[base + 2], 1); }
        if (v.w > 0.5f) { labels[b] = base + 3; atomicAdd(&counts[base + 3], 1); }
    }
}

// ---------------------------------------------------------------------------
// K3: inv_num[c] = 1 / (counts[c] + 1)
// ---------------------------------------------------------------------------
__global__ void k_inv_num(const int* __restrict__ counts,
                          float* __restrict__ inv) {
    int c = blockIdx.x * blockDim.x + threadIdx.x;
    if (c < C_) inv[c] = 1.0f / ((float)counts[c] + 1.0f);
}

// ---------------------------------------------------------------------------
// K4: gather center row, form dists, block-reduce L2 norm, scatter-add the
// SGD update into out_center with float atomics. One block (256 thr) per b.
// center gather should hit L2 (warmed by K1).
// ---------------------------------------------------------------------------
__global__ __launch_bounds__(D_)
void k_dists_norm_scatter(const float* __restrict__ x,
                          const float* __restrict__ center,
                          const int* __restrict__ labels,
                          float* __restrict__ norms,
                          float* __restrict__ out_center) {
    __shared__ float red[D_];
    int b = blockIdx.x;
    int d = threadIdx.x;
    int lab = labels[b];
    float dist = x[(size_t)b * D_ + d] - center[(size_t)lab * D_ + d];
    // new_center[c] = center[c] + LR * sum of dists over samples of class c
    atomicAdd(&out_center[(size_t)lab * D_ + d], LR_ * dist);
    red[d] = dist * dist;
    __syncthreads();
    #pragma unroll
    for (int s = D_ / 2; s > 0; s >>= 1) {
        if (d < s) red[d] += red[d + s];
        __syncthreads();
    }
    if (d == 0) norms[b] = sqrtf(red[0]);
}

// ---------------------------------------------------------------------------
// K5: loss[b,c] = norm[b] * inv_num[c]  -- rank-1 outer product, one 16x16
// tile per wave via V_WMMA_F32_16X16X4_F32 (A has norm in K=0 column, B has
// inv in K=0 row, other K slots zero). 64 x 3125 = 200000 tiles exactly,
// no bounds checks -> EXEC all ones as WMMA requires. 204.8 MB write-once
// output: non-temporal stores bypass-rinse L2 so center stays resident.
// ---------------------------------------------------------------------------
__global__ __launch_bounds__(256)
void k_loss_wmma(const float* __restrict__ norms,
                 const float* __restrict__ inv,
                 float* __restrict__ loss) {
    const int CT = C_ / 16;                      // 3125 tiles along C
    int wave = (blockIdx.x * blockDim.x + threadIdx.x) >> 5;
    int lane = threadIdx.x & 31;
    int tb = wave / CT;
    int tc = wave % CT;

    int l15 = lane & 15;
    // A 16x4 f32 layout: lanes 0-15 hold K=0 (VGPR0) / K=1 (VGPR1),
    // lanes 16-31 hold K=2/K=3. We want only K=0 populated -> mask hi lanes.
    float msk = (lane < 16) ? 1.0f : 0.0f;       // branchless, keeps EXEC full
    float nv = norms[tb * 16 + l15] * msk;
    float iv = inv[tc * 16 + l15] * msk;

    v2f a  = { nv, 0.0f };                       // A: column vector in K=0
    v2f bm = { iv, 0.0f };                       // B: row vector in K=0
    v8f c  = {};
    // (neg_a, A, neg_b, B, c_mod, C, reuse_a, reuse_b)
    v8f d = __builtin_amdgcn_wmma_f32_16x16x4_f32(
        false, a, false, bm, (short)0, c, false, false);

    // D layout: VGPR r -> row r (lanes 0-15) / row r+8 (lanes 16-31), col=lane%16
    float* outp = loss + (size_t)(tb * 16) * C_ + (size_t)tc * 16;
    int rbase = (lane < 16) ? 0 : 8;
    #pragma unroll
    for (int r = 0; r < 8; ++r)
        __builtin_nontemporal_store(d[r], &outp[(size_t)(rbase + r) * C_ + l15]);
}

// ---------------------------------------------------------------------------
extern "C" void kernel_launch(void* const* d_in, const int* in_sizes, int n_in,
                              void* d_out, int out_size, void* d_ws, size_t ws_size,
                              hipStream_t stream) {
    const float* x      = (const float*)d_in[0];   // [B, D]
    const float* y      = (const float*)d_in[1];   // [B, C] one-hot
    const float* center = (const float*)d_in[2];   // [C, D]

    float* loss       = (float*)d_out;                       // [B, C]
    float* out_center = loss + (size_t)B_ * C_;              // [C, D]

    // Workspace layout (~408 KB)
    float* norms  = (float*)d_ws;            // B_ floats
    float* inv    = norms + B_;              // C_ floats
    int*   counts = (int*)(inv + C_);        // C_ ints
    int*   labels = counts + C_;             // B_ ints

    k_zero_counts<<<(C_ + 255) / 256, 256, 0, stream>>>(counts);
    k_copy_center<<<2048, 256, 0, stream>>>((const v4f*)center,
                                            (v4f*)out_center, C_ * D_ / 4);
    k_find_labels<<<B_, 256, 0, stream>>>(y, labels, counts);
    k_inv_num<<<(C_ + 255) / 256, 256, 0, stream>>>(counts, inv);
    k_dists_norm_scatter<<<B_, D_, 0, stream>>>(x, center, labels, norms, out_center);

    // 200000 tiles, 8 waves (tiles) per 256-thread block -> 25000 blocks
    k_loss_wmma<<<(B_ / 16) * (C_ / 16) / 8, 256, 0, stream>>>(norms, inv, loss);
}